// GNNEncoder_6828998001562
// MI455X (gfx1250) — compile-verified
//
#include <hip/hip_runtime.h>
#include <cstdint>
#include <cstddef>

#define NNODES  100000
#define DIM     128
#define NEDGES  800000
#define NLAYERS 3
#define BN_EPS  1e-5f
#define ROWS    32      // node rows per workgroup in the GEMM (divides NNODES exactly)
#define PITCH   132     // LDS row pitch in floats (128 + 4 pad -> bank-conflict free)

typedef __attribute__((ext_vector_type(2))) float v2f;
typedef __attribute__((ext_vector_type(8))) float v8f;
typedef __attribute__((ext_vector_type(4))) int   i32x4;

#define HAS_ASYNC_LDS __has_builtin(__builtin_amdgcn_global_load_async_to_lds_b128)

#if HAS_ASYNC_LDS
__device__ __forceinline__ void async_copy16(const float* g, float* lds) {
    // 16B global -> LDS direct copy, tracked with ASYNCcnt (no VGPR round trip)
    __builtin_amdgcn_global_load_async_to_lds_b128(
        (__attribute__((address_space(1))) i32x4*)(uintptr_t)g,
        (__attribute__((address_space(3))) i32x4*)(uint32_t)(uintptr_t)lds,
        0, 0);
}
#endif

// ---------------------------------------------------------------- zero
__global__ void zero_kernel(float* __restrict__ p, int n) {
    int i = blockIdx.x * blockDim.x + threadIdx.x;
    int stride = gridDim.x * blockDim.x;
    for (; i < n; i += stride) p[i] = 0.0f;
}

// ---------------------------------------------------------------- edge scatter (mean-agg numerator + counts)
__global__ void scatter_kernel(const float* __restrict__ x,
                               const int* __restrict__ src,
                               const int* __restrict__ dst,
                               float* __restrict__ agg,
                               float* __restrict__ cnt, int nedges) {
    int gid  = blockIdx.x * blockDim.x + threadIdx.x;
    int e    = gid >> 5;          // one wave32 per edge
    int lane = gid & 31;
    if (e >= nedges) return;
    int s = src[e];
    int d = dst[e];
    const float4 v = ((const float4*)(x + (size_t)s * DIM))[lane];
    float* a = agg + (size_t)d * DIM + lane * 4;
    atomicAdd(a + 0, v.x);
    atomicAdd(a + 1, v.y);
    atomicAdd(a + 2, v.z);
    atomicAdd(a + 3, v.w);
    if (lane == 0) atomicAdd(cnt + d, 1.0f);
}

// ---------------------------------------------------------------- fused SAGE GEMM (+bias, +ELU, +BN partial stats)
// h = (agg/max(cnt,1)) @ Wl^T + x @ Wr^T + b ; optional ELU; accumulate per-feature sum/sumsq.
// Normalization is folded into the epilogue (diag(inv)*(agg@Wl^T)), so LDS tiles are raw copies.
__global__ __launch_bounds__(256) void sage_gemm_kernel(
    const float* __restrict__ agg, const float* __restrict__ cnt,
    const float* __restrict__ x,   const float* __restrict__ Wl,
    const float* __restrict__ Wr,  const float* __restrict__ bias,
    float* __restrict__ h, float* __restrict__ stats, int apply_elu) {

    __shared__ float sA[ROWS * PITCH];
    __shared__ float sX[ROWS * PITCH];
    __shared__ float sInv[ROWS];
    __shared__ float sSum[DIM];
    __shared__ float sSq[DIM];

    const int tid = threadIdx.x;
    const int m0  = blockIdx.x * ROWS;

    if (tid < DIM) sSum[tid] = 0.0f;
    else           sSq[tid - DIM] = 0.0f;
    if (tid < ROWS) sInv[tid] = 1.0f / fmaxf(cnt[m0 + tid], 1.0f);

#if HAS_ASYNC_LDS
    // async raw tile staging: 4 b128 per thread per tile
    for (int i = tid; i < ROWS * (DIM / 4); i += 256) {
        int row = i >> 5;          // DIM/4 == 32 float4 per row
        int c4  = i & 31;
        int m   = m0 + row;
        async_copy16(agg + (size_t)m * DIM + c4 * 4, sA + row * PITCH + c4 * 4);
        async_copy16(x   + (size_t)m * DIM + c4 * 4, sX + row * PITCH + c4 * 4);
    }
    asm volatile("s_wait_asynccnt 0x0" ::: "memory");
#else
    for (int i = tid; i < ROWS * (DIM / 4); i += 256) {
        int row = i >> 5;
        int c4  = i & 31;
        int m   = m0 + row;
        float4 av = ((const float4*)(agg + (size_t)m * DIM))[c4];
        float4 xv = ((const float4*)(x + (size_t)m * DIM))[c4];
        *(float4*)(sA + row * PITCH + c4 * 4) = av;
        *(float4*)(sX + row * PITCH + c4 * 4) = xv;
    }
#endif
    __syncthreads();

    const int wave = tid >> 5;
    const int lane = tid & 31;
    const int hi   = lane >> 4;    // half-wave selects K pair per 16x16x4 layout
    const int rc   = lane & 15;    // A-frag row == B/C-frag column
    const int n    = wave * 16 + rc;

    const float* wlrow = Wl + (size_t)n * DIM + hi * 2;
    const float* wrrow = Wr + (size_t)n * DIM + hi * 2;
    const float  bv    = bias[n];

    for (int r = 0; r < ROWS / 16; ++r) {
        // two independent accumulator chains (agg branch / root branch)
        v8f ca = {0.f, 0.f, 0.f, 0.f, 0.f, 0.f, 0.f, 0.f};
        v8f cr = {0.f, 0.f, 0.f, 0.f, 0.f, 0.f, 0.f, 0.f};

        const float* arow = sA + (r * 16 + rc) * PITCH + hi * 2;
        const float* xrow = sX + (r * 16 + rc) * PITCH + hi * 2;
#pragma unroll
        for (int kk = 0; kk < 32; ++kk) {
            v2f a, bw, ax, bx;
            a.x  = arow[kk * 4];  a.y  = arow[kk * 4 + 1];
            bw.x = wlrow[kk * 4]; bw.y = wlrow[kk * 4 + 1];
            ca = __builtin_amdgcn_wmma_f32_16x16x4_f32(false, a, false, bw,
                                                       (short)0, ca, false, false);
            ax.x = xrow[kk * 4];  ax.y = xrow[kk * 4 + 1];
            bx.x = wrrow[kk * 4]; bx.y = wrrow[kk * 4 + 1];
            cr = __builtin_amdgcn_wmma_f32_16x16x4_f32(false, ax, false, bx,
                                                       (short)0, cr, false, false);
        }

        // epilogue: mean-normalize agg branch, bias, ELU, store h, BN partial stats
        float ls = 0.0f, lq = 0.0f;
#pragma unroll
        for (int v = 0; v < 8; ++v) {
            int ridx = r * 16 + v + hi * 8;     // C layout: VGPR v -> M = v + 8*hi
            float val = sInv[ridx] * ca[v] + cr[v] + bv;
            if (apply_elu) val = (val > 0.0f) ? val : expm1f(val);
            h[(size_t)(m0 + ridx) * DIM + n] = val;
            ls += val;
            lq += val * val;
        }
        atomicAdd(&sSum[n], ls);
        atomicAdd(&sSq[n], lq);
    }
    __syncthreads();
    if (tid < DIM) atomicAdd(&stats[tid], sSum[tid]);
    else           atomicAdd(&stats[tid], sSq[tid - DIM]);
}

// ---------------------------------------------------------------- BN stats -> scale/shift
__global__ void bn_finalize_kernel(const float* __restrict__ stats,
                                   const float* __restrict__ gamma,
                                   const float* __restrict__ beta,
                                   float* __restrict__ ss) {
    int d = threadIdx.x;
    if (d >= DIM) return;
    float mean = stats[d] * (1.0f / (float)NNODES);
    float var  = stats[DIM + d] * (1.0f / (float)NNODES) - mean * mean;
    var = fmaxf(var, 0.0f);
    float sc = gamma[d] * rsqrtf(var + BN_EPS);
    ss[d]       = sc;
    ss[DIM + d] = beta[d] - mean * sc;
}

// ---------------------------------------------------------------- BN apply
__global__ void bn_apply_kernel(const float* __restrict__ h,
                                const float* __restrict__ ss,
                                float* __restrict__ out) {
    int i = blockIdx.x * blockDim.x + threadIdx.x;
    int stride = gridDim.x * blockDim.x;
    const int total = NNODES * (DIM / 4);
    for (; i < total; i += stride) {
        int c4 = i & 31;                       // column quad (DIM/4 == 32)
        float4 hv = ((const float4*)h)[i];
        float4 sc = ((const float4*)ss)[c4];
        float4 sh = ((const float4*)(ss + DIM))[c4];
        float4 o;
        o.x = sc.x * hv.x + sh.x;
        o.y = sc.y * hv.y + sh.y;
        o.z = sc.z * hv.z + sh.z;
        o.w = sc.w * hv.w + sh.w;
        ((float4*)out)[i] = o;
    }
}

// ---------------------------------------------------------------- launch
extern "C" void kernel_launch(void* const* d_in, const int* in_sizes, int n_in,
                              void* d_out, int out_size, void* d_ws, size_t ws_size,
                              hipStream_t stream) {
    const float* x0    = (const float*)d_in[0];
    const int*   ei    = (const int*)d_in[1];
    const float* Wl    = (const float*)d_in[2];
    const float* Wr    = (const float*)d_in[3];
    const float* b     = (const float*)d_in[4];
    const float* gamma = (const float*)d_in[5];
    const float* beta  = (const float*)d_in[6];
    const int* src = ei;
    const int* dst = ei + NEDGES;

    // workspace layout: [agg | cnt | stats] contiguous (one zero pass), then ss, h, xbuf
    float* agg   = (float*)d_ws;
    float* cnt   = agg + (size_t)NNODES * DIM;
    float* stats = cnt + NNODES;
    float* ss    = stats + 256;                 // 256-float scale/shift
    float* hbuf  = ss + 256;
    float* xbuf  = hbuf + (size_t)NNODES * DIM;

    const int ZN = NNODES * DIM + NNODES + 256; // agg + cnt + stats

    for (int l = 0; l < NLAYERS; ++l) {
        const float* xin  = (l == 0) ? x0 : xbuf;
        float*       xout = (l == NLAYERS - 1) ? (float*)d_out : xbuf;
        const float* wl   = Wl + (size_t)l * DIM * DIM;
        const float* wr   = Wr + (size_t)l * DIM * DIM;

        zero_kernel<<<2048, 256, 0, stream>>>(agg, ZN);
        scatter_kernel<<<(NEDGES * 32 + 255) / 256, 256, 0, stream>>>(
            xin, src, dst, agg, cnt, NEDGES);
        sage_gemm_kernel<<<NNODES / ROWS, 256, 0, stream>>>(
            agg, cnt, xin, wl, wr, b + l * DIM, hbuf, stats,
            (l < NLAYERS - 1) ? 1 : 0);
        bn_finalize_kernel<<<1, 128, 0, stream>>>(stats, gamma + l * DIM,
                                                  beta + l * DIM, ss);
        bn_apply_kernel<<<1024, 256, 0, stream>>>(hbuf, ss, xout);
    }
}